// ScoreHead_50757923504886
// MI455X (gfx1250) — compile-verified
//
#include <hip/hip_runtime.h>
#include <math.h>

// ---- problem constants (from reference) ----
#define EMBED   768
#define NHEADS  8
#define HEADDIM 96
#define BQ      256
#define NTOT    16384
#define SCALE_F 0.1020620726159658f   // 96^-0.5

typedef __attribute__((ext_vector_type(2))) float    v2f;
typedef __attribute__((ext_vector_type(4))) float    v4f;
typedef __attribute__((ext_vector_type(8))) float    v8f;
typedef __attribute__((ext_vector_type(4))) unsigned v4u;
typedef __attribute__((ext_vector_type(8))) int      v8i;
typedef __attribute__((ext_vector_type(4))) int      v4i;

// D = A(16x4) * B(4x16) + C, full fp32 WMMA (CDNA5 V_WMMA_F32_16X16X4_F32).
// Per-lane fragment mapping (ISA 7.12.2): half = lane>>4, lr = lane&15
//   A: vgpr v holds A[row=lr][k = 2*half + v]        (contiguous pair -> b64)
//   B: mirror:  vgpr v holds B[k = 2*half + v][col=lr]
//   D: vgpr r holds D[m = r + 8*half][n = lr]
static __device__ __forceinline__ v8f wmma_f32(v2f a, v2f b, v8f c) {
    return __builtin_amdgcn_wmma_f32_16x16x4_f32(false, a, false, b,
                                                 (short)0, c, false, false);
}

// -------------------------------------------------------------------------
// Kernel 1: Y[m, o] = sum_i X[m,i] * W[o,i] + bias[o]     (K = N_out = 768)
// Block: 256 threads = 8 waves, tile 128(M) x 128(O); wave tile 32x64
// (2 A-frags x 4 B-frags -> 8 WMMAs per 6 b64 loads, ~21 FLOP/B from L2).
// -------------------------------------------------------------------------
__global__ __launch_bounds__(256)
void proj_gemm(const float* __restrict__ X, const float* __restrict__ W,
               const float* __restrict__ bias, float* __restrict__ Y)
{
    const int lane = threadIdx.x & 31;
    const int wave = threadIdx.x >> 5;
    const int half = lane >> 4;
    const int lr   = lane & 15;
    const int wm   = wave & 3;     // 4 waves along M
    const int wn   = wave >> 2;    // 2 waves along O
    const int m0   = blockIdx.x * 128 + wm * 32;
    const int n0   = blockIdx.y * 128 + wn * 64;

    v8f acc[2][4] = {};   // [mi][oj] 16x16 tiles

    const float* xp[2];
    const float* wp[4];
    #pragma unroll
    for (int i = 0; i < 2; ++i)
        xp[i] = X + (size_t)(m0 + 16 * i + lr) * EMBED + 2 * half;
    #pragma unroll
    for (int j = 0; j < 4; ++j)
        wp[j] = W + (size_t)(n0 + 16 * j + lr) * EMBED + 2 * half;

    #pragma unroll 4
    for (int kk = 0; kk < EMBED; kk += 4) {
        v2f a[2], b[4];
        #pragma unroll
        for (int i = 0; i < 2; ++i) a[i] = *(const v2f*)(xp[i] + kk);
        #pragma unroll
        for (int j = 0; j < 4; ++j) b[j] = *(const v2f*)(wp[j] + kk);
        #pragma unroll
        for (int i = 0; i < 2; ++i)
            #pragma unroll
            for (int j = 0; j < 4; ++j)
                acc[i][j] = wmma_f32(a[i], b[j], acc[i][j]);
    }

    float bj[4];
    #pragma unroll
    for (int j = 0; j < 4; ++j) bj[j] = bias[n0 + 16 * j + lr];

    #pragma unroll
    for (int i = 0; i < 2; ++i) {
        #pragma unroll
        for (int r = 0; r < 8; ++r) {
            const int row = m0 + 16 * i + r + 8 * half;
            #pragma unroll
            for (int j = 0; j < 4; ++j)
                Y[(size_t)row * EMBED + n0 + 16 * j + lr] = acc[i][j][r] + bj[j];
        }
    }
}

// -------------------------------------------------------------------------
// Kernel 2: per-head logits + head-MLP fusion.
// Block: 16 b-rows x 128 n-cols, 8 waves; each wave owns one 16x16 (b,n)
// tile and 8 rotating accumulators (one per head = 24 WMMA K-steps).
// Q tile (16x768 f32) is staged into LDS by the Tensor Data Mover with
// pad_enable: 4 DWORDs of padding after every 256 DWORDs -> LDS row stride
// 780 floats (12 mod 64 -> conflict-free b64 ds_loads). Element (r,c) lives
// at r*780 + c + 4*(c>>8). K rows stream from global (L2-resident, 50 MB
// fits the 192 MB L2). Each lane ends holding all 8 head-logits of its
// (b,n) elements, so the H->4->GELU->1 MLP is lane-local VALU.
// -------------------------------------------------------------------------
#define QS_STRIDE 780   // 768 + 3 pads x 4 DWORDs (TDM pad layout)

#if !defined(__clang_major__) || __clang_major__ < 23
#define USE_TDM 1       // ROCm 7.2 / clang-22: 5-arg tensor_load_to_lds
#endif

__global__ __launch_bounds__(256)
void score_kernel(const float* __restrict__ QP, const float* __restrict__ KP,
                  const float* __restrict__ W1, const float* __restrict__ b1,
                  const float* __restrict__ W2, const float* __restrict__ b2,
                  float* __restrict__ out)
{
    __shared__ float qs[16 * QS_STRIDE];

    const int tid  = threadIdx.x;
    const int lane = tid & 31;
    const int wave = tid >> 5;
    const int half = lane >> 4;
    const int lr   = lane & 15;
    const int b0   = blockIdx.x * 16;
    const int n0   = blockIdx.y * 128 + wave * 16;

#ifdef USE_TDM
    // ---- Tensor Data Mover: 2-D tile 768x16 f32, QP row-major, padded LDS.
    if (wave == 0) {
        const unsigned lds_byte = (unsigned)(uintptr_t)(&qs[0]);  // generic low32 == LDS offset
        const unsigned long long gaddr =
            (unsigned long long)(uintptr_t)(QP + (size_t)b0 * EMBED);
        // D# group0: count=1 | lds_addr | global_addr | type=2
        v4u g0 = { 1u,
                   lds_byte,
                   (unsigned)(gaddr & 0xFFFFFFFFull),
                   (unsigned)(gaddr >> 32) | (2u << 30) };
        // D# group1: data_size=4B(2), pad_enable, pad_interval=256dw(7),
        // pad_amount=4dw(3); tensor_dim0=768, tensor_dim1=256;
        // tile_dim0=768, tile_dim1=16; tensor_dim0_stride=768.
        v8i g1 = { 0x07D20000,
                   (int)((EMBED & 0xFFFF) << 16),   // tensor_dim0 lo16 @ [63:48]
                   (int)((BQ & 0xFFFF) << 16),      // tensor_dim1 lo16 @ [95:80]
                   (int)((EMBED & 0xFFFF) << 16),   // tile_dim0     @ [127:112]
                   16,                              // tile_dim1     @ [143:128]
                   EMBED,                           // dim0_stride lo32
                   0, 0 };
        v4i gz = { 0, 0, 0, 0 };                    // groups 2/3 unused (2-D)
        __builtin_amdgcn_tensor_load_to_lds(g0, g1, gz, gz, 0);
        __builtin_amdgcn_s_wait_tensorcnt(0);
    }
#else
    // ---- Fallback: cooperative b128 copy into the same padded layout.
    for (int i = tid; i < (16 * EMBED) / 4; i += 256) {
        const int e = i * 4;
        const int r = e / EMBED, c = e % EMBED;
        const int cc = c + 4 * (c >> 8);            // TDM-equivalent padding
        *(v4f*)(&qs[r * QS_STRIDE + cc]) =
            *(const v4f*)(&QP[(size_t)(b0 + r) * EMBED + c]);
    }
#endif
    __syncthreads();

    v8f acc[NHEADS] = {};

    const float* qrow = qs + lr * QS_STRIDE + 2 * half;               // LDS
    const float* krow = KP + (size_t)(n0 + lr) * EMBED + 2 * half;    // L2

    #pragma unroll
    for (int h = 0; h < NHEADS; ++h) {
        v8f a_ = acc[h];
        #pragma unroll 8
        for (int t = 0; t < HEADDIM / 4; ++t) {
            const int kk  = h * HEADDIM + t * 4;
            const int kof = kk + 4 * (kk >> 8);     // padded LDS column
            v2f av = *(const v2f*)(qrow + kof);     // ds_load_b64
            v2f bv = *(const v2f*)(krow + kk);      // global_load_b64 (L2 hit)
            a_ = wmma_f32(av, bv, a_);
        }
        acc[h] = a_;
    }

    // Head-fusion MLP (scale folded into W1): z = (l*S)@W1^T + b1,
    // g = exact GELU, s = g@W2^T + b2.   All lane-local.
    float w1[4][NHEADS], bb1[4], w2[4];
    #pragma unroll
    for (int j = 0; j < 4; ++j) {
        bb1[j] = b1[j];
        w2[j]  = W2[j];
        #pragma unroll
        for (int h = 0; h < NHEADS; ++h) w1[j][h] = W1[j * NHEADS + h] * SCALE_F;
    }
    const float bb2 = b2[0];

    #pragma unroll
    for (int r = 0; r < 8; ++r) {
        float s = bb2;
        #pragma unroll
        for (int j = 0; j < 4; ++j) {
            float z = bb1[j];
            #pragma unroll
            for (int h = 0; h < NHEADS; ++h) z += w1[j][h] * acc[h][r];
            const float g = 0.5f * z * (1.0f + erff(z * 0.70710678118654752f));
            s += w2[j] * g;
        }
        out[(size_t)(b0 + r + 8 * half) * NTOT + (n0 + lr)] = s;
    }
}

// -------------------------------------------------------------------------
extern "C" void kernel_launch(void* const* d_in, const int* in_sizes, int n_in,
                              void* d_out, int out_size, void* d_ws, size_t ws_size,
                              hipStream_t stream)
{
    (void)in_sizes; (void)n_in; (void)out_size; (void)ws_size;
    const float* query  = (const float*)d_in[0];
    const float* search = (const float*)d_in[1];
    const float* Wq     = (const float*)d_in[2];
    const float* bq     = (const float*)d_in[3];
    const float* Wk     = (const float*)d_in[4];
    const float* bk     = (const float*)d_in[5];
    const float* W1     = (const float*)d_in[6];
    const float* b1     = (const float*)d_in[7];
    const float* W2     = (const float*)d_in[8];
    const float* b2     = (const float*)d_in[9];
    float* out = (float*)d_out;

    float* qp = (float*)d_ws;                       // 256   x 768
    float* kp = qp + (size_t)BQ * EMBED;            // 16384 x 768  (~51 MB total)

    proj_gemm<<<dim3(BQ   / 128, EMBED / 128), 256, 0, stream>>>(query,  Wq, bq, qp);
    proj_gemm<<<dim3(NTOT / 128, EMBED / 128), 256, 0, stream>>>(search, Wk, bk, kp);
    score_kernel<<<dim3(BQ / 16, NTOT / 128), 256, 0, stream>>>(qp, kp, W1, b1, W2, b2, out);
}